// BasisEncoder_25890062860681
// MI455X (gfx1250) — compile-verified
//
#include <hip/hip_runtime.h>
#include <stdint.h>

// One-hot encoder, store-bandwidth-bound (256 MB fp32 out, ~11.5us floor at
// 23.3 TB/s HBM). Strategy: zero-fill a 64KB LDS tile with conflict-free
// b128 stores, scatter one 1.0f per row, then DMA each wave's 8KB sub-tile
// to global memory with the CDNA5 Tensor Data Mover (tensor_store_from_lds,
// non-temporal: output > 192MB L2 and is never re-read on device).

typedef uint32_t v4u __attribute__((ext_vector_type(4)));
typedef uint32_t v8u __attribute__((ext_vector_type(8)));

#define ROWS_PER_BLOCK 256
#define THREADS 256

__global__ __launch_bounds__(THREADS) void BasisEncoder_tdm_kernel(
    const int* __restrict__ x, float* __restrict__ out, int batch) {
  __shared__ float4 tile[ROWS_PER_BLOCK * 64 / 4];  // 64 KB: 256 rows x 64 f32

  const int t = threadIdx.x;
  const int r0 = blockIdx.x * ROWS_PER_BLOCK;

  // Each thread owns output row (r0 + t); fetch its x (coalesced b32 load).
  int row = r0 + t;
  int xv = (row < batch) ? x[row] : 0;

  // Zero-fill the tile in linear float4 order: consecutive lanes hit
  // consecutive 16B chunks -> bank-conflict-free ds_store_b128, zero VALU.
  const float4 z = {0.0f, 0.0f, 0.0f, 0.0f};
#pragma unroll
  for (int j = 0; j < 16; ++j) {
    tile[j * 256 + t] = z;  // 16 stores, immediate offsets j*4096 off one base
  }
  __syncthreads();  // zeros from all waves visible before per-row scatter

  // Scatter the hot element: (x%256)%64 == x&63 for x>=0. Thread t writes
  // into row t, which lies inside its own wave's TDM sub-tile, so only an
  // in-wave dscnt wait is needed before the DMA (no second barrier).
  unsigned idx = ((unsigned)xv) & 63u;
  ((float*)tile)[(unsigned)t * 64u + idx] = 1.0f;
  asm volatile("s_wait_dscnt 0x0" ::: "memory");

  // Each of the 8 waves DMA-stores its own disjoint 8KB sub-tile (32 rows).
  // Tensor ops ignore EXEC, so descriptors are per-wave disjoint (no
  // divergent-branch skipping relied upon); wave id forced scalar.
  unsigned wave = __builtin_amdgcn_readfirstlane((unsigned)t >> 5);

  // LDS byte offset of the tile: flat-pointer low 32 bits are the LDS offset.
  uint32_t lds_off = (uint32_t)(uintptr_t)&tile[0] + wave * 8192u;
  uint64_t ga = (uint64_t)(uintptr_t)out +
                (uint64_t)blockIdx.x * (uint64_t)(ROWS_PER_BLOCK * 64 * 4) +
                (uint64_t)wave * 8192ull;

  // ---- D# group 0 (128b): count=1 | lds_addr | global_addr[56:0] | type=2
  v4u g0;
  g0.x = 1u;                   // [1:0] count=1, user descriptor
  g0.y = lds_off;              // [63:32] lds_addr (bytes)
  g0.z = (uint32_t)ga;         // [95:64] global_addr[31:0]
  g0.w = ((uint32_t)(ga >> 32) & 0x01FFFFFFu) | (2u << 30);  // addr[56:32]|type=2

  // ---- D# group 1 (256b): 1-D tile of 2048 x 4-byte elements (8 KB)
  v8u g1;
  g1[0] = 0x00020000u;         // workgroup_mask=0, data_size=2 (4B), no extras
  g1[1] = 0x0800u << 16;       // tensor_dim0[15:0]=2048 at bits[63:48]
  g1[2] = 1u << 16;            // tensor_dim0 hi=0; tensor_dim1[15:0]=1
  g1[3] = 0x0800u << 16;       // tensor_dim1 hi=0; tile_dim0=2048 at [127:112]
  g1[4] = 0u;                  // tile_dim1=0 (unused), tile_dim2=0
  g1[5] = 0x0800u;             // tensor_dim0_stride=2048 (low 32)
  g1[6] = 0u;                  // stride0 hi=0; tensor_dim1_stride lo=0
  g1[7] = 0u;                  // tensor_dim1_stride hi=0

  asm volatile("tensor_store_from_lds %0, %1 th:TH_STORE_NT"
               :
               : "s"(g0), "s"(g1)
               : "memory");

  // Don't let the workgroup release LDS while the TDM is still reading it.
  __builtin_amdgcn_s_wait_tensorcnt(0);
}

extern "C" void kernel_launch(void* const* d_in, const int* in_sizes, int n_in,
                              void* d_out, int out_size, void* d_ws,
                              size_t ws_size, hipStream_t stream) {
  const int* x = (const int*)d_in[0];
  float* out = (float*)d_out;
  int batch = in_sizes[0];  // 1048576, a multiple of 256
  int blocks = (batch + ROWS_PER_BLOCK - 1) / ROWS_PER_BLOCK;
  BasisEncoder_tdm_kernel<<<blocks, THREADS, 0, stream>>>(x, out, batch);
}